// Attention_53764400611491
// MI455X (gfx1250) — compile-verified
//
#include <hip/hip_runtime.h>

// Problem constants (from reference): b=2, C=64, h=w=32 -> N=1024 spatial.
// d = C/HEADS = 1 collapses attention to o = q * (k . v per channel), so the
// whole module is 4 small GEMMs + a per-channel dot + a diagonal scale.
#define CCH 64
#define NSP 1024
#define NB  2
#define NTILES (NSP / 16)

typedef __attribute__((ext_vector_type(2))) float v2f;
typedef __attribute__((ext_vector_type(8))) float v8f;

// ---- WMMA f32 16x16x4 fragments (ISA 7.12.2 layouts, wave32) ----

// A 16x4 (MxK) from row-major W[64][64]: lanes 0-15 -> {K0,K1}, 16-31 -> {K2,K3}.
// k is always even -> 8-byte aligned b64 load.
__device__ __forceinline__ v2f load_a_frag(const float* __restrict__ W,
                                           int m0, int k0, int lane) {
  const int m = m0 + (lane & 15);
  const int k = k0 + ((lane >> 4) << 1);
  return *reinterpret_cast<const v2f*>(W + m * CCH + k);
}

// B 4x16 (KxN) from an LDS tile xs[c][n] (row stride 16 words).
// Lane halves hit disjoint 16-bank quartets -> conflict-free ds reads.
__device__ __forceinline__ v2f load_b_lds(const float* __restrict__ xs,
                                          int k0, int lane) {
  const int n = lane & 15;
  const int k = k0 + ((lane >> 4) << 1);
  v2f b;
  b.x = xs[k * 16 + n];
  b.y = xs[(k + 1) * 16 + n];
  return b;
}

// ---- Kernel 1: Q = Wq@X (to ws); Spart[b,tile,c] = sum_{n in tile} K*V ----
// grid: (NTILES, NB); block: 128 = 4 waves, wave w owns M rows [16w,16w+16).
__global__ __launch_bounds__(128) void qkv_s_kernel(
    const float* __restrict__ x,  const float* __restrict__ wq,
    const float* __restrict__ wk, const float* __restrict__ wv,
    float* __restrict__ Q, float* __restrict__ Spart) {
  __shared__ float xs[CCH * 16];  // 4 KB X tile, shared by all 3 GEMMs
  const int tid  = threadIdx.x;
  const int lane = tid & 31;
  const int n0   = blockIdx.x << 4;
  const int b    = blockIdx.y;
  const int m0   = (tid >> 5) << 4;
  const float* Xb = x + b * CCH * NSP;

  // Stage X[:, n0:n0+16] -> LDS with coalesced float4 loads (64B per row).
  for (int i = tid; i < CCH * 4; i += 128) {
    const int c = i >> 2, j = i & 3;
    reinterpret_cast<float4*>(xs)[i] =
        *reinterpret_cast<const float4*>(Xb + c * NSP + n0 + 4 * j);
  }
  __syncthreads();

  // Three accumulators share each B fragment (one LDS read, three WMMAs).
  v8f q = {}, kk = {}, vv = {};
#pragma unroll
  for (int k = 0; k < CCH; k += 4) {
    const v2f bb = load_b_lds(xs, k, lane);
    q  = __builtin_amdgcn_wmma_f32_16x16x4_f32(
        false, load_a_frag(wq, m0, k, lane), false, bb, (short)0, q,  false, false);
    kk = __builtin_amdgcn_wmma_f32_16x16x4_f32(
        false, load_a_frag(wk, m0, k, lane), false, bb, (short)0, kk, false, false);
    vv = __builtin_amdgcn_wmma_f32_16x16x4_f32(
        false, load_a_frag(wv, m0, k, lane), false, bb, (short)0, vv, false, false);
  }

  // Q tile -> workspace (half-wave writes 64B contiguous per row).
  float* Qb = Q + b * CCH * NSP;
  const int nn = n0 + (lane & 15);
  const int mh = (lane >> 4) << 3;
#pragma unroll
  for (int i = 0; i < 8; ++i) Qb[(m0 + i + mh) * NSP + nn] = q[i];

  // Per-channel K.V dot over this n-tile; reduce across 16 N-lanes per half.
  float p[8];
#pragma unroll
  for (int i = 0; i < 8; ++i) p[i] = kk[i] * vv[i];
#pragma unroll
  for (int off = 1; off < 16; off <<= 1) {
#pragma unroll
    for (int i = 0; i < 8; ++i) p[i] += __shfl_xor(p[i], off, 32);
  }
  if ((lane & 15) == 0) {
#pragma unroll
    for (int i = 0; i < 8; ++i)
      Spart[(b * NTILES + blockIdx.x) * CCH + (m0 + i + mh)] = p[i];
  }
}

// ---- Kernel 2: Out = Wp @ (diag(s) * Q) + X,  s = sum of Spart tiles ----
__global__ __launch_bounds__(128) void out_kernel(
    const float* __restrict__ x, const float* __restrict__ wp,
    const float* __restrict__ Q, const float* __restrict__ Spart,
    float* __restrict__ out) {
  __shared__ float xs[CCH * 16];
  __shared__ float sl[CCH];
  const int tid  = threadIdx.x;
  const int lane = tid & 31;
  const int n0   = blockIdx.x << 4;
  const int b    = blockIdx.y;
  const int m0   = (tid >> 5) << 4;

  // Reduce the 64 tile-partials per channel (coalesced over c; L2-resident).
  if (tid < CCH) {
    float s = 0.0f;
    const float* Sp = Spart + b * NTILES * CCH + tid;
    for (int t = 0; t < NTILES; ++t) s += Sp[t * CCH];
    sl[tid] = s;
  }
  __syncthreads();

  // Stage diag(s) * Q[:, n0:n0+16] -> LDS, folding the scale into staging.
  const float* Qb = Q + b * CCH * NSP;
  for (int i = tid; i < CCH * 4; i += 128) {
    const int c = i >> 2, j = i & 3;
    float4 v = *reinterpret_cast<const float4*>(Qb + c * NSP + n0 + 4 * j);
    const float s = sl[c];
    v.x *= s; v.y *= s; v.z *= s; v.w *= s;
    reinterpret_cast<float4*>(xs)[i] = v;
  }
  __syncthreads();

  v8f acc = {};
#pragma unroll
  for (int k = 0; k < CCH; k += 4) {
    acc = __builtin_amdgcn_wmma_f32_16x16x4_f32(
        false, load_a_frag(wp, m0, k, lane), false, load_b_lds(xs, k, lane),
        (short)0, acc, false, false);
  }

  const float* Xb = x + b * CCH * NSP;
  float* Ob = out + b * CCH * NSP;
  const int nn = n0 + (lane & 15);
  const int mh = (lane >> 4) << 3;
#pragma unroll
  for (int i = 0; i < 8; ++i) {
    const int m = m0 + i + mh;
    Ob[m * NSP + nn] = acc[i] + Xb[m * NSP + nn];
  }
}

extern "C" void kernel_launch(void* const* d_in, const int* in_sizes, int n_in,
                              void* d_out, int out_size, void* d_ws, size_t ws_size,
                              hipStream_t stream) {
  const float* x  = (const float*)d_in[0];
  const float* wq = (const float*)d_in[1];
  const float* wk = (const float*)d_in[2];
  const float* wv = (const float*)d_in[3];
  const float* wp = (const float*)d_in[4];
  float* out = (float*)d_out;

  // Workspace: Q[2][64][1024] (512 KB) then Spart[2][64][64] (32 KB).
  float* Q     = (float*)d_ws;
  float* Spart = Q + NB * CCH * NSP;

  dim3 grid(NTILES, NB);
  qkv_s_kernel<<<grid, dim3(128), 0, stream>>>(x, wq, wk, wv, Q, Spart);
  out_kernel<<<grid, dim3(128), 0, stream>>>(x, wp, Q, Spart, out);
}